// DOnePoleCell_37641093382426
// MI455X (gfx1250) — compile-verified
//
#include <hip/hip_runtime.h>

typedef float v2f __attribute__((ext_vector_type(2)));
typedef float v8f __attribute__((ext_vector_type(8)));

#define T_STEPS 16384
#define B_COLS  4096

__device__ __forceinline__ float clamp1(float v) {
    return fminf(fmaxf(v, -1.0f), 1.0f);
}

// Entry of the 16x16 lower-triangular tile operator:
// M[i][j] = b0 (i==j), c*a^(i-1-j) (j<i), 0 (j>i)
__device__ __forceinline__ float mentry(int i, int j, float b0, float c, float a) {
    if (j > i) return 0.0f;
    if (j == i) return b0;
    float r = c;
    int e = i - 1 - j;            // 0..14
#pragma unroll
    for (int k = 0; k < 14; ++k) r = (k < e) ? r * a : r;
    return r;
}

// ---------------------------------------------------------------------------
// Kernel 1: per-(chunk,col) carry = chunk-end state assuming s0 = 0.
// Fully coalesced stream of x; one thread per (chunk,col).
// ---------------------------------------------------------------------------
__global__ void iir_carry_kernel(const float* __restrict__ x,
                                 const float* __restrict__ b0p,
                                 const float* __restrict__ b1p,
                                 const float* __restrict__ a1p,
                                 float* __restrict__ hbuf,
                                 int LC) {
    int gid   = blockIdx.x * blockDim.x + threadIdx.x;   // chunk*B + col
    int col   = gid & (B_COLS - 1);
    int chunk = gid >> 12;                               // log2(4096)

    float a = clamp1(a1p[0]);
    float c = fmaf(a, b0p[0], b1p[0]);                   // c = b1 + a*b0

    const float* xp = x + (size_t)chunk * LC * B_COLS + col;
    float s = 0.0f;
#pragma unroll 8
    for (int t = 0; t < LC; ++t) {
        s = fmaf(a, s, c * xp[(size_t)t * B_COLS]);
    }
    hbuf[gid] = s;
}

// ---------------------------------------------------------------------------
// Kernel 2: per-column scan over NC chunk carries with multiplier a^LC.
// Emits each chunk's true initial state and the final state output.
// ---------------------------------------------------------------------------
__global__ void iir_scan_kernel(const float* __restrict__ state,
                                const float* __restrict__ a1p,
                                const float* __restrict__ hbuf,
                                float* __restrict__ s0buf,
                                float* __restrict__ final_state,
                                int NC, int LC) {
    int col = blockIdx.x * blockDim.x + threadIdx.x;
    if (col >= B_COLS) return;

    float a = clamp1(a1p[0]);
    // aLC = a^LC via square-and-multiply
    float aLC = 1.0f, base = a;
    int e = LC;
    while (e) { if (e & 1) aLC *= base; base *= base; e >>= 1; }

    float s = state[col];
    for (int k = 0; k < NC; ++k) {
        s0buf[(size_t)k * B_COLS + col] = s;
        s = fmaf(aLC, s, hbuf[(size_t)k * B_COLS + col]);
    }
    final_state[col] = s;
}

// ---------------------------------------------------------------------------
// Kernel 3: output pass. One wave per (chunk, 32-column tile pair). Each
// 16-step time tile is Out = M·X + a^i ⊗ s0 via 4 chained
// V_WMMA_F32_16X16X4_F32 per 16-col sub-tile; the two sub-tiles form
// independent wmma chains that interleave (no RAW dead cycles). State hops
// tiles with a single ds_bpermute per sub-tile (FMA folded pre-shuffle).
// ---------------------------------------------------------------------------
__global__ void iir_wmma_out_kernel(const float* __restrict__ x,
                                    const float* __restrict__ b0p,
                                    const float* __restrict__ b1p,
                                    const float* __restrict__ a1p,
                                    const float* __restrict__ s0buf,
                                    float* __restrict__ out,
                                    int LC) {
    const int lane    = threadIdx.x & 31;
    const int h       = lane >> 4;        // half-wave (0/1)
    const int cl      = lane & 15;        // column within sub-tile / A row
    const int wid     = blockIdx.x * (blockDim.x >> 5) + (threadIdx.x >> 5);
    const int PAIRS   = B_COLS / 32;      // 128 column tile-pairs
    const int chunk   = wid / PAIRS;
    const int colbase = (wid % PAIRS) * 32;

    const float b0 = b0p[0];
    const float b1 = b1p[0];
    const float a  = clamp1(a1p[0]);
    const float c  = fmaf(a, b0, b1);

    // A operand: M split into four 16x4 blocks, CDNA5 f32 A layout:
    // VGPR0 holds K = 2h, VGPR1 holds K = 2h+1 (K relative to block).
    v2f Ablk[4];
#pragma unroll
    for (int q = 0; q < 4; ++q) {
        int j0 = 4 * q + 2 * h;
        Ablk[q].x = mentry(cl, j0,     b0, c, a);
        Ablk[q].y = mentry(cl, j0 + 1, b0, c, a);
    }

    // a^(r + 8h) for accumulator seeding (C/D row = r + 8h)
    const float a2 = a * a, a4 = a2 * a2, a8 = a4 * a4;
    float pwl[8];
    {
        float p = h ? a8 : 1.0f;
#pragma unroll
        for (int r = 0; r < 8; ++r) { pwl[r] = p; p *= a; }
    }

    // True initial states of this chunk for the two 16-col sub-tiles
    const float* s0p = s0buf + (size_t)chunk * B_COLS + colbase + cl;
    float s0 = s0p[0];
    float s1 = s0p[16];

    const int t0base = chunk * LC;
    const int ntiles = LC / 16;
    for (int tt = 0; tt < ntiles; ++tt) {
        const int t0 = t0base + tt * 16;
        const float* xp0 = x + (size_t)t0 * B_COLS + colbase + cl;
        const float* xp1 = xp0 + 16;

        // B operands: X rows 4q..4q+3 (VGPR0: K=2h, VGPR1: K=2h+1)
        v2f B0[4], B1[4];
#pragma unroll
        for (int q = 0; q < 4; ++q) {
            size_t tr = (size_t)(4 * q + 2 * h) * B_COLS;
            B0[q].x = xp0[tr];
            B0[q].y = xp0[tr + B_COLS];
            B1[q].x = xp1[tr];
            B1[q].y = xp1[tr + B_COLS];
        }

        // Seed accumulators with the homogeneous term a^row * s
        v8f acc0, acc1;
#pragma unroll
        for (int r = 0; r < 8; ++r) {
            acc0[r] = pwl[r] * s0;
            acc1[r] = pwl[r] * s1;
        }

        // Two independent 4-deep WMMA chains (interleave; no RAW stalls)
#pragma unroll
        for (int q = 0; q < 4; ++q) {
            acc0 = __builtin_amdgcn_wmma_f32_16x16x4_f32(
                false, Ablk[q], false, B0[q], (short)0, acc0, false, false);
            acc1 = __builtin_amdgcn_wmma_f32_16x16x4_f32(
                false, Ablk[q], false, B1[q], (short)0, acc1, false, false);
        }

        // Store both 16x16 output tiles (row = r + 8h)
        float* op0 = out + (size_t)t0 * B_COLS + colbase + cl;
        float* op1 = op0 + 16;
#pragma unroll
        for (int r = 0; r < 8; ++r) {
            size_t ro = (size_t)(r + 8 * h) * B_COLS;
            op0[ro] = acc0[r];
            op1[ro] = acc1[r];
        }

        // Next tile state: s' = a*Out[15] + b1*X[15]. Row 15 lives in the
        // upper half (lane 16+cl); fold the FMA before a single shuffle.
        float v0 = fmaf(a, acc0[7], b1 * B0[3].y);
        float v1 = fmaf(a, acc1[7], b1 * B1[3].y);
        s0 = __shfl(v0, 16 + cl, 32);
        s1 = __shfl(v1, 16 + cl, 32);
    }
}

// ---------------------------------------------------------------------------
extern "C" void kernel_launch(void* const* d_in, const int* in_sizes, int n_in,
                              void* d_out, int out_size, void* d_ws, size_t ws_size,
                              hipStream_t stream) {
    const float* x     = (const float*)d_in[0];
    const float* state = (const float*)d_in[1];
    const float* b0    = (const float*)d_in[2];
    const float* b1    = (const float*)d_in[3];
    const float* a1    = (const float*)d_in[4];
    float*       out   = (float*)d_out;

    // Choose chunk count to fit 2*NC*B floats of scratch (carries + s0)
    int NC = 128;
    while (NC > 1 && (size_t)2 * NC * B_COLS * sizeof(float) > ws_size) NC >>= 1;
    const int LC = T_STEPS / NC;

    float* hbuf  = (float*)d_ws;
    float* s0buf = hbuf + (size_t)NC * B_COLS;

    // Pass 1: chunk carries (s0 = 0)
    {
        const int threads = 256;
        const int total   = NC * B_COLS;
        iir_carry_kernel<<<total / threads, threads, 0, stream>>>(
            x, b0, b1, a1, hbuf, LC);
    }

    // Pass 2: inter-chunk scan + final-state output (appended after T*B outputs)
    iir_scan_kernel<<<B_COLS / 256, 256, 0, stream>>>(
        state, a1, hbuf, s0buf, out + (size_t)T_STEPS * B_COLS, NC, LC);

    // Pass 3: WMMA output pass, one wave per (chunk, 32-column tile pair)
    {
        const int threads    = 256;                 // 8 waves per block
        const int totalWaves = NC * (B_COLS / 32);  // NC*128, divisible by 8
        iir_wmma_out_kernel<<<totalWaves / 8, threads, 0, stream>>>(
            x, b0, b1, a1, s0buf, out, LC);
    }
}